// FeatureBlock_8383776161799
// MI455X (gfx1250) — compile-verified
//
#include <hip/hip_runtime.h>
#include <math.h>

// MI455X / gfx1250: wave32, WMMA f32 16x16x4 for the window reductions.
// Problem is HBM-bound (8 MB essential traffic, ~0.35us @ 23.3 TB/s);
// one block per batch row stages the 16KB channel row into LDS once.

typedef __attribute__((ext_vector_type(2))) float v2f;
typedef __attribute__((ext_vector_type(8))) float v8f;

#define M_CH   3
#define DIL    4
#define EPS_F  1e-8f
#define C_CH   8
#define Q_LEN  4096
#define L_LEN  32
#define W_LEN  3972          // Q - (L-1)*DIL
#define NTILES 249           // ceil(W_LEN / 16)

__global__ __launch_bounds__(256)
void shapelet_min_kernel(const float* __restrict__ x,
                         const float* __restrict__ shp,
                         const float* __restrict__ posMap,
                         const float* __restrict__ posSlope,
                         float* __restrict__ out)
{
    __shared__ __align__(16) float xb[4112];  // 4096 + padding for tail-tile reads
    __shared__ float wb[3984];                // positional weights
    __shared__ float red[8];                  // per-wave partial mins

    const int tid  = threadIdx.x;
    const int b    = blockIdx.x;
    const int lane = tid & 31;
    const int wave = tid >> 5;
    const int half = lane >> 4;   // 0: lanes 0-15, 1: lanes 16-31
    const int mn   = lane & 15;   // row/col index within 16

    // ---- stage x[b, M_CH, :] into LDS (16KB, coalesced float4) ----
    const float4* xrow = (const float4*)(x + ((size_t)b * C_CH + M_CH) * Q_LEN);
    float4* xb4 = (float4*)xb;
#pragma unroll
    for (int i = 0; i < 4; ++i)
        xb4[tid + 256 * i] = xrow[tid + 256 * i];

    // ---- positional weights: w = slope*elu(-posMap) + 2 (once per block) ----
    const float slope = posSlope[0];
    for (int i = tid; i < W_LEN; i += 256) {
        float z = -posMap[i];
        float e = (z > 0.f) ? z : expm1f(z);
        wb[i] = slope * e + 2.0f;
    }

    // ---- shapelet stats + constant A-matrix fragments (filters) ----
    // A (16x4 per chunk): row 0 = shapelet, row 1 = ones, rows 2-15 = zero.
    // f32 A layout: lane holds M = lane%16; K = vgpr + 2*(lane/16)  (+4*chunk).
    float sum_s = 0.f, sq_s = 0.f;
#pragma unroll
    for (int k = 0; k < L_LEN; ++k) {
        float sv = shp[k];          // uniform -> scalar loads
        sum_s += sv;
        sq_s  += sv * sv;
    }
    float a0[8], a1[8];
#pragma unroll
    for (int c = 0; c < 8; ++c) {
        float s0 = shp[4 * c + 2 * half];
        float s1 = shp[4 * c + 2 * half + 1];
        a0[c] = (mn == 0) ? s0 : ((mn == 1) ? 1.0f : 0.0f);
        a1[c] = (mn == 0) ? s1 : ((mn == 1) ? 1.0f : 0.0f);
    }

    __syncthreads();

    // ---- per-wave tile loop: 16 windows per tile, wave-uniform control ----
    float mymin = INFINITY;
    for (int t = wave; t < NTILES; t += 8) {
        const int w0 = t * 16;
        v8f dx = {};   // D[m][n]: m=filter, n=window  (x matmul)
        v8f dq = {};   // same with x^2 (for sumsq)
#pragma unroll
        for (int c = 0; c < 8; ++c) {
            // B (4x16): lane holds N = lane%16, K = vgpr + 2*(lane/16) (+4*chunk)
            // element B[k][n] = x[w0 + n + DIL*(4c + k)]
            const int base = w0 + mn + 16 * c + 8 * half;
            float b0 = xb[base];
            float b1 = xb[base + 4];
            v2f av = {a0[c], a1[c]};
            v2f bv = {b0, b1};
            v2f bq = {b0 * b0, b1 * b1};
            dx = __builtin_amdgcn_wmma_f32_16x16x4_f32(
                    false, av, false, bv, (short)0, dx, false, false);
            dq = __builtin_amdgcn_wmma_f32_16x16x4_f32(
                    false, av, false, bq, (short)0, dq, false, false);
        }
        // lanes 0-15: c[0] = row m=0 (dot), c[1] = row m=1 (sum / sumsq)
        float dot = dx[0];
        float sm  = dx[1];
        float sq  = dq[1];
        int   w   = w0 + mn;

        float mu    = sm * (1.0f / 32.0f);
        float M2    = fmaxf(sq - sm * mu, 0.0f);        // sum (x-mu)^2
        float sd    = sqrtf(M2 * (1.0f / 31.0f));       // ddof=1 std
        float inv   = 1.0f / (sd + EPS_F);
        float cross = (dot - mu * sum_s) * inv;         // sum s*xn
        float xn2   = M2 * inv * inv;                   // sum xn^2
        float d     = (sq_s - 2.0f * cross + xn2) * (1.0f / 32.0f);

        int   wi = (w < W_LEN) ? w : (W_LEN - 1);
        float f  = d * wb[wi];
        bool  valid = (half == 0) && (w < W_LEN);
        mymin = fminf(mymin, valid ? f : INFINITY);
    }

    // ---- reductions: lanes -> wave, waves -> block ----
#pragma unroll
    for (int off = 16; off > 0; off >>= 1)
        mymin = fminf(mymin, __shfl_xor(mymin, off, 32));
    if (lane == 0) red[wave] = mymin;
    __syncthreads();
    if (tid == 0) {
        float a = red[0];
#pragma unroll
        for (int i = 1; i < 8; ++i) a = fminf(a, red[i]);
        out[b] = fmaxf(a, 0.0f);   // relu
    }
}

extern "C" void kernel_launch(void* const* d_in, const int* in_sizes, int n_in,
                              void* d_out, int out_size, void* d_ws, size_t ws_size,
                              hipStream_t stream) {
    const float* x        = (const float*)d_in[0];   // (512, 8, 4096) f32
    const float* shapelet = (const float*)d_in[1];   // (32,) f32
    const float* posMap   = (const float*)d_in[2];   // (3972,) f32
    const float* posSlope = (const float*)d_in[3];   // scalar f32
    (void)d_in[4];                                   // ep (unused)
    float* out = (float*)d_out;                      // (512,) f32

    shapelet_min_kernel<<<512, 256, 0, stream>>>(x, shapelet, posMap, posSlope, out);
}